// Graph_Tanh_W_78477642432810
// MI455X (gfx1250) — compile-verified
//
#include <hip/hip_runtime.h>
#include <math.h>

#define N_NODES 8192
#define K_SEL   64
#define THREADS 256
#define CHUNKS  8                  // float4 chunks per thread (8*4 = 32 elems)
#define EPT     32                 // elements per thread

typedef float  f32x4 __attribute__((ext_vector_type(4)));
typedef unsigned u32x4 __attribute__((ext_vector_type(4)));

__device__ __forceinline__ float fast_tanh(float x) {
#if __has_builtin(__builtin_amdgcn_tanhf)
    return __builtin_amdgcn_tanhf(x);            // gfx1250 v_tanh_f32
#elif __has_builtin(__builtin_amdgcn_tanh_f32)
    return __builtin_amdgcn_tanh_f32(x);
#else
    return tanhf(x);
#endif
}

__device__ __forceinline__ void wait_async0() {
#if __has_builtin(__builtin_amdgcn_s_wait_asynccnt)
    __builtin_amdgcn_s_wait_asynccnt(0);
#else
    asm volatile("s_wait_asynccnt 0" ::: "memory");
#endif
}

__global__ __launch_bounds__(THREADS) void topk_tanh_row_kernel(
    const float* __restrict__ A, float* __restrict__ Out)
{
    __shared__ float    buf[N_NODES];       // 32 KB row staging (async target)
    __shared__ unsigned hist[8 * 256];      // per-wave privatized histograms
    __shared__ unsigned bc_bin, bc_kth, bc_tie;

    const int tid  = threadIdx.x;
    const int wave = tid >> 5;
    const int lane = tid & 31;
    // Low 32 bits of a shared-aperture flat address == LDS byte offset.
    const unsigned bufBase = (unsigned)(size_t)(&buf[0]);

    // Async copy of one 32KB row into buf: 8 waves x 32 lanes x 8 B128 each.
    auto issue_row_load = [&](const float* rowPtr) {
        unsigned long long base = (unsigned long long)rowPtr;
        unsigned off = (unsigned)(wave * 4096 + lane * 16);
#pragma unroll
        for (int c = 0; c < 8; ++c) {
            unsigned ldsOff = bufBase + off + (unsigned)(c * 512);
            unsigned gOff   = off + (unsigned)(c * 512);
            asm volatile("global_load_async_to_lds_b128 %0, %1, %2 th:TH_LOAD_NT"
                         :: "v"(ldsOff), "v"(gOff), "s"(base)
                         : "memory");
        }
    };

    int row = (int)blockIdx.x;
    const int stride = (int)gridDim.x;
    if (row < N_NODES) issue_row_load(A + (size_t)row * N_NODES);

    for (; row < N_NODES; row += stride) {
        // ---- wait for this row's async copy (own wave) + all waves ----
        wait_async0();
        __syncthreads();

        // ---- pull my 32 elements into registers ----
        unsigned vb[EPT];
        const f32x4* buf4 = (const f32x4*)buf;
#pragma unroll
        for (int c = 0; c < CHUNKS; ++c) {
            f32x4 f = buf4[c * 256 + tid];
            vb[c * 4 + 0] = __float_as_uint(f.x);
            vb[c * 4 + 1] = __float_as_uint(f.y);
            vb[c * 4 + 2] = __float_as_uint(f.z);
            vb[c * 4 + 3] = __float_as_uint(f.w);
        }
        __syncthreads();   // everyone finished reading buf

        // ---- prefetch next row: overlaps the whole selection phase ----
        int nextRow = row + stride;
        if (nextRow < N_NODES) issue_row_load(A + (size_t)nextRow * N_NODES);

        // ---- radix-select the 64th-largest |a| (monotonic in bits) ----
        unsigned kth = K_SEL;      // rank still sought among matching elems
        unsigned prefHi = 0;       // resolved high bits (right-aligned)
#pragma unroll
        for (int p = 0; p < 4; ++p) {
            const int shift = 24 - 8 * p;
            // zero histograms: 2048 u32 = 512 u32x4, 2 per thread
            {
                u32x4* h4 = (u32x4*)hist;
                u32x4 z = (u32x4)(0u);
                h4[tid] = z;
                h4[tid + 256] = z;
            }
            __syncthreads();

            unsigned* h = &hist[wave * 256];
#pragma unroll
            for (int e = 0; e < EPT; ++e) {
                unsigned u = vb[e] & 0x7fffffffu;
                bool match = (p == 0) || ((u >> (shift + 8)) == prefHi);
                if (match) atomicAdd(&h[(u >> shift) & 255u], 1u);
            }
            __syncthreads();

            // wave 0: fold 8 copies + 256-bin suffix scan, no wg barriers
            if (wave == 0) {
                const int base = lane * 8;
                unsigned c0[8] = {0, 0, 0, 0, 0, 0, 0, 0};
                const u32x4* h4 = (const u32x4*)hist;
#pragma unroll
                for (int w = 0; w < 8; ++w) {
                    u32x4 a = h4[(w * 256 + base) >> 2];
                    u32x4 b = h4[((w * 256 + base) >> 2) + 1];
                    c0[0] += a.x; c0[1] += a.y; c0[2] += a.z; c0[3] += a.w;
                    c0[4] += b.x; c0[5] += b.y; c0[6] += b.z; c0[7] += b.w;
                }
                // local (in-register) suffix sums over my 8 bins
                unsigned ls[8];
                ls[7] = c0[7];
#pragma unroll
                for (int j = 6; j >= 0; --j) ls[j] = ls[j + 1] + c0[j];
                unsigned S = ls[0];
                // wave32 inclusive suffix scan across lanes (Hillis-Steele)
                unsigned T = S;
#pragma unroll
                for (int off = 1; off < 32; off <<= 1) {
                    unsigned t = __shfl(T, lane + off, 32);
                    if (lane + off < 32) T += t;
                }
                unsigned Texcl = T - S;   // suffix starting at lane+1
#pragma unroll
                for (int j = 0; j < 8; ++j) {
                    unsigned hi = ls[j] + Texcl;                 // suffix(bin)
                    unsigned lo = ((j < 7) ? ls[j + 1] : 0u) + Texcl;
                    if (hi >= kth && lo < kth) {   // unique crossing
                        bc_bin = (unsigned)(base + j);
                        bc_kth = kth - lo;
                    }
                }
            }
            __syncthreads();
            prefHi = (prefHi << 8) | bc_bin;
            kth    = bc_kth;
            // next write to bc_* is >=2 barriers away; no extra barrier needed
        }

        const unsigned T = prefHi;     // exact bits of the K-th largest |a|
        if (tid == 0) bc_tie = 0u;
        __syncthreads();

        // ---- emit: tanh(3a) for selected, 0 otherwise; NT b128 stores ----
        f32x4* out4 = (f32x4*)(Out + (size_t)row * N_NODES);
#pragma unroll
        for (int c = 0; c < CHUNKS; ++c) {
            f32x4 o;
#pragma unroll
            for (int q = 0; q < 4; ++q) {
                unsigned bits = vb[c * 4 + q];
                unsigned u    = bits & 0x7fffffffu;
                bool sel = (u > T);
                if (!sel && u == T) {
                    unsigned slot = atomicAdd(&bc_tie, 1u);
                    sel = (slot < kth);
                }
                float val = 0.0f;
                if (sel) val = fast_tanh(3.0f * __uint_as_float(bits));
                o[q] = val;
            }
            __builtin_nontemporal_store(o, &out4[c * 256 + tid]);
        }
        // next iteration's wait+barrier orders bc_tie / buf reuse
    }
}

extern "C" void kernel_launch(void* const* d_in, const int* in_sizes, int n_in,
                              void* d_out, int out_size, void* d_ws, size_t ws_size,
                              hipStream_t stream) {
    // setup_inputs order: idx (int64, unused), A (float32 [N,N])
    const float* A = (const float*)d_in[1];
    float* Out = (float*)d_out;
    (void)in_sizes; (void)n_in; (void)out_size; (void)d_ws; (void)ws_size;

    dim3 grid(1024), block(THREADS);
    hipLaunchKernelGGL(topk_tanh_row_kernel, grid, block, 0, stream, A, Out);
}